// BiGNN_17626545783660
// MI455X (gfx1250) — compile-verified
//
#include <hip/hip_runtime.h>

typedef __attribute__((ext_vector_type(2))) float v2f;
typedef __attribute__((ext_vector_type(8))) float v8f;

#define D 128
#define LDS_STRIDE 132   // 128 + 4 pad: rows land on distinct LDS banks
#define MTILE 64         // rows per block (4 x 16 M-subtiles)

// ---------------------------------------------------------------------------
// Kernel 1: zero the x = L@E accumulator (must happen every launch).
// ---------------------------------------------------------------------------
__global__ __launch_bounds__(256) void bignn_zero(float* __restrict__ p, int n4) {
    int i = blockIdx.x * 256 + threadIdx.x;
    if (i < n4) ((float4*)p)[i] = make_float4(0.f, 0.f, 0.f, 0.f);
}

// ---------------------------------------------------------------------------
// Kernel 2: COO SpMM  x[rows[e],:] += vals[e] * features[cols[e],:]
// One wave32 per edge: e is wave-uniform -> scalar loads for row/col/val.
// Each lane covers 4 consecutive floats (float4 gather, 4 hw f32 atomics,
// L2-resident since features = 51 MB < 192 MB L2).
// ---------------------------------------------------------------------------
__global__ __launch_bounds__(256) void bignn_spmm(const int*   __restrict__ rows,
                                                  const int*   __restrict__ cols,
                                                  const float* __restrict__ vals,
                                                  const float* __restrict__ feat,
                                                  float*       __restrict__ x,
                                                  int E) {
    int t = blockIdx.x * 256 + threadIdx.x;
    int e = t >> 5;                 // wave-uniform
    if (e >= E) return;
    int lane = threadIdx.x & 31;

    int   c = cols[e];
    int   r = rows[e];
    float v = vals[e];

    const float4 f = *(const float4*)(feat + (size_t)c * D + lane * 4);
    float* dst = x + (size_t)r * D + lane * 4;
    unsafeAtomicAdd(dst + 0, v * f.x);
    unsafeAtomicAdd(dst + 1, v * f.y);
    unsafeAtomicAdd(dst + 2, v * f.z);
    unsafeAtomicAdd(dst + 3, v * f.w);
}

// ---------------------------------------------------------------------------
// Kernel 3: out = (f + x) @ W1 + (x * f) @ W2 + (b1 + b2)
// Block: 256 thr = 8 waves -> one 64(M) x 128(N) output slab.
// Each wave: 4 M-subtiles x one 16-wide N-tile, 4 accumulators, B fragment
// loaded once per k-step and reused 4x (quarters W L2 traffic, 4 WMMA per
// B load, and round-robins accumulators so back-to-back WMMAs never chain
// on the same D register).
//
// fp32 WMMA operand layout (ISA 7.12.2, 32-bit A 16x4 / B 4x16 / C 16x16):
//   A: lane<16 holds row M=lane,  VGPR0/1 = K=0,1 ; lane>=16 -> K=2,3
//   B: lane<16 holds col N=lane,  VGPR0/1 = K=0,1 ; lane>=16 -> K=2,3
//   C: VGPR i: lanes0-15 -> (M=i, N=lane), lanes16-31 -> (M=i+8, N=lane-16)
// ---------------------------------------------------------------------------
__global__ __launch_bounds__(256) void bignn_gemm(const float* __restrict__ feat,
                                                  const float* __restrict__ x,
                                                  const float* __restrict__ W1,
                                                  const float* __restrict__ b1,
                                                  const float* __restrict__ W2,
                                                  const float* __restrict__ b2,
                                                  float*       __restrict__ out,
                                                  int N) {
    __shared__ float h1[MTILE * LDS_STRIDE];
    __shared__ float h2[MTILE * LDS_STRIDE];

    const int node0 = blockIdx.x * MTILE;

    // Stage h1 = f + x, h2 = f * x for 64 rows into LDS (2048 float4 chunks).
    for (int i = threadIdx.x; i < MTILE * 32; i += 256) {
        int rrow = i >> 5;              // 0..63
        int c4   = (i & 31) << 2;       // 0,4,...,124
        int node = node0 + rrow;
        float4 f  = make_float4(0.f, 0.f, 0.f, 0.f);
        float4 xv = f;
        if (node < N) {
            size_t g = (size_t)node * D + c4;
            f  = *(const float4*)(feat + g);
            xv = *(const float4*)(x + g);
        }
        int l = rrow * LDS_STRIDE + c4;
        h1[l + 0] = f.x + xv.x;  h2[l + 0] = f.x * xv.x;
        h1[l + 1] = f.y + xv.y;  h2[l + 1] = f.y * xv.y;
        h1[l + 2] = f.z + xv.z;  h2[l + 2] = f.z * xv.z;
        h1[l + 3] = f.w + xv.w;  h2[l + 3] = f.w * xv.w;
    }
    __syncthreads();

    const int wave = threadIdx.x >> 5;      // 0..7 -> N-tile
    const int lane = threadIdx.x & 31;
    const int half = lane >> 4;             // 0: K=0,1 ; 1: K=2,3
    const int l    = lane & 15;             // A row / B col within tile
    const int nc   = wave * 16 + l;         // global output column

    v8f acc[4] = {};

    // GEMM 1: acc[m] += h1[m-subtile] @ W1   (B fragment reused 4x)
    #pragma unroll
    for (int k = 0; k < D; k += 4) {
        const int k0 = k + half * 2;
        v2f b;
        b.x = W1[(size_t)k0 * D + nc];
        b.y = W1[(size_t)(k0 + 1) * D + nc];
        #pragma unroll
        for (int m = 0; m < 4; ++m) {
            v2f a = *(const v2f*)(&h1[(m * 16 + l) * LDS_STRIDE + k0]);
            acc[m] = __builtin_amdgcn_wmma_f32_16x16x4_f32(false, a, false, b,
                                                           (short)0, acc[m],
                                                           false, false);
        }
    }
    // GEMM 2 (same accumulators): acc[m] += h2[m-subtile] @ W2
    #pragma unroll
    for (int k = 0; k < D; k += 4) {
        const int k0 = k + half * 2;
        v2f b;
        b.x = W2[(size_t)k0 * D + nc];
        b.y = W2[(size_t)(k0 + 1) * D + nc];
        #pragma unroll
        for (int m = 0; m < 4; ++m) {
            v2f a = *(const v2f*)(&h2[(m * 16 + l) * LDS_STRIDE + k0]);
            acc[m] = __builtin_amdgcn_wmma_f32_16x16x4_f32(false, a, false, b,
                                                           (short)0, acc[m],
                                                           false, false);
        }
    }

    const float bias = b1[nc] + b2[nc];
    #pragma unroll
    for (int m = 0; m < 4; ++m) {
        #pragma unroll
        for (int i = 0; i < 8; ++i) {
            int node = node0 + m * 16 + half * 8 + i;
            if (node < N) out[(size_t)node * D + nc] = acc[m][i] + bias;
        }
    }
}

// ---------------------------------------------------------------------------
extern "C" void kernel_launch(void* const* d_in, const int* in_sizes, int n_in,
                              void* d_out, int out_size, void* d_ws, size_t ws_size,
                              hipStream_t stream) {
    const int*   rows = (const int*)  d_in[0];
    const int*   cols = (const int*)  d_in[1];
    const float* vals = (const float*)d_in[2];
    const float* feat = (const float*)d_in[3];
    const float* W1   = (const float*)d_in[4];
    const float* b1   = (const float*)d_in[5];
    const float* W2   = (const float*)d_in[6];
    const float* b2   = (const float*)d_in[7];
    float* out = (float*)d_out;
    float* x   = (float*)d_ws;          // [N, D] accumulator

    const int E = in_sizes[0];
    const int N = in_sizes[3] / D;

    const int n4 = (N * D) / 4;
    hipLaunchKernelGGL(bignn_zero, dim3((n4 + 255) / 256), dim3(256), 0, stream,
                       x, n4);

    const long long spmm_threads = (long long)E * 32;
    hipLaunchKernelGGL(bignn_spmm,
                       dim3((unsigned)((spmm_threads + 255) / 256)), dim3(256), 0, stream,
                       rows, cols, vals, feat, x, E);

    hipLaunchKernelGGL(bignn_gemm, dim3((N + MTILE - 1) / MTILE), dim3(256), 0, stream,
                       feat, x, W1, b1, W2, b2, out, N);
}